// DetectionLayer_8220567404574
// MI455X (gfx1250) — compile-verified
//
#include <hip/hip_runtime.h>
#include <hip/hip_bf16.h>

// ---------------- problem constants (from reference) ----------------
#define BS    4
#define FH    37
#define FW    62
#define NA    9
#define NN    (FH * FW * NA)   // 20646 anchors per batch
#define NPAD  32768            // pow2 padding for bitonic sort
#define PPRE  6000             // KEEP_PRE
#define POST  300              // KEEP_POST
#define PT    376              // padded 16-wide column tiles (375 real + 1 zero pad)
#define IOU_C (0.7f / 1.7f)    // thr/(1+thr): iou>thr <=> inter > c*(areaI+areaJ)

typedef float v2f __attribute__((ext_vector_type(2)));
typedef float v8f __attribute__((ext_vector_type(8)));
typedef unsigned int u32x4 __attribute__((ext_vector_type(4)));
typedef unsigned int u32x8 __attribute__((ext_vector_type(8)));

__device__ __forceinline__ unsigned ballot32(bool p) {
#if __has_builtin(__builtin_amdgcn_ballot_w32)
  return __builtin_amdgcn_ballot_w32(p);
#else
  return (unsigned)__ballot(p);
#endif
}

// ---------------- kernel 1: decode boxes, scores, sort keys ----------------
__global__ __launch_bounds__(256) void decode_kernel(
    const float* __restrict__ preds, const float* __restrict__ regs,
    const int* __restrict__ img_h, const int* __restrict__ img_w,
    float* __restrict__ boxes_all, float* __restrict__ scores_all,
    unsigned long long* __restrict__ keys) {
  int t = blockIdx.x * blockDim.x + threadIdx.x;
  if (t >= BS * NPAD) return;
  int b = t / NPAD, p = t % NPAD;
  unsigned long long key = ~0ull;  // padding sorts last (ascending sort)
  if (p < NN) {
    const float scl[3] = {8.f, 16.f, 32.f};   // STRIDE * SCALES
    const float rat[3] = {0.5f, 1.f, 2.f};
    int a = p % NA, cell = p / NA;
    int xq = cell % FW, yq = cell / FW;
    int ri = a / 3, si = a % 3;
    float s = scl[si], r = rat[ri];
    float w = s * sqrtf(1.0f / r);
    float h = s * sqrtf(r);
    float cx = (float)xq * 16.0f;
    float cy = (float)yq * 16.0f;
    size_t ri4 = ((size_t)b * NN + p) * 4;
    float d0 = regs[ri4 + 0], d1 = regs[ri4 + 1];
    float d2 = regs[ri4 + 2], d3 = regs[ri4 + 3];
    float pcx = d0 * w + cx;
    float pcy = d1 * h + cy;
    float pw = __expf(d2) * w;
    float ph = __expf(d3) * h;
    float W = (float)img_w[0], H = (float)img_h[0];
    float4 bx;
    bx.x = fminf(fmaxf(pcx - 0.5f * pw, 0.f), W);
    bx.y = fminf(fmaxf(pcy - 0.5f * ph, 0.f), H);
    bx.z = fminf(fmaxf(pcx + 0.5f * pw, 0.f), W);
    bx.w = fminf(fmaxf(pcy + 0.5f * ph, 0.f), H);
    reinterpret_cast<float4*>(boxes_all)[(size_t)b * NN + p] = bx;
    float sc = 1.0f / (1.0f + __expf(-preds[(size_t)b * NN + p]));
    scores_all[(size_t)b * NN + p] = sc;
    // key: ascending sort == descending score, ties -> ascending index (stable)
    unsigned u = __float_as_uint(sc);
    unsigned ord = u ^ ((u >> 31) ? 0xFFFFFFFFu : 0x80000000u);
    key = ((unsigned long long)(~ord) << 32) | (unsigned)p;
  }
  keys[t] = key;
}

// ---------------- kernel 2: one bitonic compare-exchange stage ----------------
__global__ __launch_bounds__(256) void bitonic_kernel(
    unsigned long long* __restrict__ keys, int k, int j) {
  int t = blockIdx.x * blockDim.x + threadIdx.x;
  if (t >= BS * NPAD) return;
  int b = t / NPAD, i = t % NPAD;
  int ixj = i ^ j;
  if (ixj > i) {
    unsigned long long* seg = keys + (size_t)b * NPAD;
    unsigned long long a = seg[i], c = seg[ixj];
    bool asc = (i & k) == 0;
    if ((a > c) == asc) { seg[i] = c; seg[ixj] = a; }
  }
}

// ---------------- kernel 3: gather top-P, precompute scaled areas ----------------
__global__ __launch_bounds__(256) void gather_kernel(
    const unsigned long long* __restrict__ keys,
    const float* __restrict__ boxes_all, const float* __restrict__ scores_all,
    float* __restrict__ sbox, float* __restrict__ sscore, float* __restrict__ sA) {
  int t = blockIdx.x * blockDim.x + threadIdx.x;
  if (t >= BS * PPRE) return;
  int b = t / PPRE;
  unsigned idx = (unsigned)(keys[(size_t)b * NPAD + (t % PPRE)] & 0xFFFFFFFFull);
  if (idx >= NN) idx = NN - 1;  // cannot happen (N > P) — safety clamp
  float4 bx = reinterpret_cast<const float4*>(boxes_all)[(size_t)b * NN + idx];
  reinterpret_cast<float4*>(sbox)[t] = bx;
  sscore[t] = scores_all[(size_t)b * NN + idx];
  float area = fmaxf(bx.z - bx.x, 0.f) * fmaxf(bx.w - bx.y, 0.f);
  sA[t] = IOU_C * area;
}

// ---------------- kernel 4: suppression bitmask (TDM + async LDS + WMMA) ----------------
// grid = (375 row tiles, 47 col stripes of 8 waves, BS); block = 256 (8 waves)
__global__ __launch_bounds__(256) void nms_mask_kernel(
    const float* __restrict__ sbox, const float* __restrict__ sA,
    unsigned short* __restrict__ mask) {
  const int b = blockIdx.z;
  const int rowTile = blockIdx.x;              // 0..374
  const int wave = threadIdx.x >> 5;           // 0..7
  const int lane = threadIdx.x & 31;
  const int ct = blockIdx.y * 8 + wave;        // 0..375 (375 = zero pad tile)
  const int rowbase = rowTile * 16;
  const int colbase = ct * 16;

  const float4* bb4 = reinterpret_cast<const float4*>(sbox) + (size_t)b * PPRE;
  const float* aa = sA + (size_t)b * PPRE;

  __shared__ float4 rowbox[16];
  __shared__ float rowSA[16];

  if (wave == 0) {
    // --- Tensor Data Mover: DMA the 16 row boxes (64 f32 = 256B) global->LDS.
    // D# group0: count=1 | lds_addr | global_addr[56:0] | type=2.
    // D# group1: data_size=4B, tensor_dim0=tile_dim0=64, tensor_dim1=tile_dim1=1,
    //            tensor_dim0_stride=64, workgroup_mask=0 (not in a cluster).
    unsigned ldsb = (unsigned)(size_t)(void*)&rowbox[0];
    unsigned long long ga =
        (unsigned long long)(size_t)(const void*)(bb4 + rowbase);
    u32x4 g0;
    g0.x = 1u;                                            // count=1, user mode
    g0.y = ldsb;                                          // lds_addr
    g0.z = (unsigned)(ga & 0xFFFFFFFFull);                // global_addr lo
    g0.w = (unsigned)((ga >> 32) & 0x01FFFFFFull) | 0x80000000u;  // hi | type=2
    u32x8 g1;
    g1[0] = 0x00020000u;        // workgroup_mask=0, data_size=2 (4 bytes)
    g1[1] = 64u << 16;          // tensor_dim0[15:0]=64 (bits 63:48)
    g1[2] = 1u << 16;           // tensor_dim0[31:16]=0, tensor_dim1[15:0]=1
    g1[3] = 64u << 16;          // tensor_dim1[31:16]=0, tile_dim0=64
    g1[4] = 1u;                 // tile_dim1=1, tile_dim2=0
    g1[5] = 64u;                // tensor_dim0_stride[31:0]=64
    g1[6] = 0u;                 // stride hi, tensor_dim1_stride lo
    g1[7] = 0u;                 // tensor_dim1_stride hi
    asm volatile("tensor_load_to_lds %0, %1" :: "s"(g0), "s"(g1) : "memory");

    // --- Async global->LDS for the 16 scaled row areas (per-lane b32 DMA).
    if (lane < 16) {
      const float* gs = aa + (rowbase + lane);
      unsigned ldss = (unsigned)(size_t)(void*)&rowSA[lane];
      asm volatile("global_load_async_to_lds_b32 %0, %1, off"
                   :: "v"(ldss), "v"(gs) : "memory");
    }
    asm volatile("s_wait_asynccnt 0" ::: "memory");
#if __has_builtin(__builtin_amdgcn_s_wait_tensorcnt)
    __builtin_amdgcn_s_wait_tensorcnt(0);
#else
    asm volatile("s_wait_tensorcnt 0" ::: "memory");
#endif
  }
  __syncthreads();

  // Column box / scaled area for this lane (clamped; pad tile masked via pred).
  const int gj = colbase + (lane & 15);
  const int jc = (gj < PPRE) ? gj : (PPRE - 1);
  float4 bj = bb4[jc];
  float sAj = aa[jc];

  // D[m][n] = sA_row[m] + sA_col[n] via v_wmma_f32_16x16x4_f32:
  //   A(16x4): col0 = sA_row, col1 = ones;  B(4x16): row0 = ones, row1 = sA_col.
  const bool lo = lane < 16;
  v8f D;
#if __has_builtin(__builtin_amdgcn_wmma_f32_16x16x4_f32)
  {
    v2f A, B;
    A.x = lo ? rowSA[lane & 15] : 0.0f;  // K=0 (lanes 0-15) / K=2 (zeros)
    A.y = lo ? 1.0f : 0.0f;              // K=1 ones        / K=3 zeros
    B.x = lo ? 1.0f : 0.0f;              // row K=0: ones
    B.y = lo ? sAj : 0.0f;               // row K=1: sA_col
    v8f C = {0.f, 0.f, 0.f, 0.f, 0.f, 0.f, 0.f, 0.f};
    D = __builtin_amdgcn_wmma_f32_16x16x4_f32(false, A, false, B, (short)0, C,
                                              false, false);
  }
#else
  {
    const int mb = (lane >> 4) << 3;
#pragma unroll
    for (int v = 0; v < 8; ++v) D[v] = rowSA[mb + v] + sAj;
  }
#endif

  // Per-lane: 8 pair tests matching the WMMA C/D layout; ballot -> row masks.
  const int mbase = (lane >> 4) << 3;  // lanes 0-15 -> rows v, lanes 16-31 -> rows v+8
  unsigned bal[8];
#pragma unroll
  for (int v = 0; v < 8; ++v) {
    int m = mbase + v;
    float4 bi = rowbox[m];
    float xx1 = fmaxf(bi.x, bj.x);
    float yy1 = fmaxf(bi.y, bj.y);
    float xx2 = fminf(bi.z, bj.z);
    float yy2 = fminf(bi.w, bj.w);
    float inter = fmaxf(xx2 - xx1, 0.f) * fmaxf(yy2 - yy1, 0.f);
    int gi = rowbase + m;
    bool pred = (gj < PPRE) && (gj > gi) && (inter > D[v]);
    bal[v] = ballot32(pred);
  }
  if (lane == 0) {
#pragma unroll
    for (int v = 0; v < 8; ++v) {
      size_t r0 = (size_t)b * PPRE + (rowbase + v);
      size_t r1 = (size_t)b * PPRE + (rowbase + v + 8);
      mask[r0 * PT + ct] = (unsigned short)(bal[v] & 0xFFFFu);
      mask[r1 * PT + ct] = (unsigned short)(bal[v] >> 16);
    }
  }
}

// ---------------- kernel 5: serial greedy sweep + output ----------------
__global__ __launch_bounds__(256) void nms_sweep_kernel(
    const unsigned short* __restrict__ mask, const float* __restrict__ sbox,
    const float* __restrict__ sscore, float* __restrict__ out) {
  const int b = blockIdx.x;
  const int t = threadIdx.x;
  const unsigned* mrow =
      reinterpret_cast<const unsigned*>(mask + (size_t)b * PPRE * PT);
  const int W = PT / 2;  // 188 u32 words per row

  __shared__ unsigned rem[PT / 2];
  __shared__ int kidx[POST];
  __shared__ int kc;

  for (int w = t; w < W; w += blockDim.x) rem[w] = 0u;
  if (t == 0) kc = 0;
  __syncthreads();

  for (int i = 0; i < PPRE; ++i) {
    if (kc >= POST) break;  // remaining kept boxes can't reach the output
    bool kept = ((rem[i >> 5] >> (i & 31)) & 1u) == 0u;
    __syncthreads();
    if (kept) {  // uniform across block
      if (t == 0) kidx[kc] = i;
      const unsigned* row = mrow + (size_t)i * W;
      for (int w = t; w < W; w += blockDim.x) rem[w] |= row[w];
      __syncthreads();
      if (t == 0) kc += 1;
    }
    __syncthreads();
  }
  __syncthreads();

  // Write [POST,4] for this batch; zero rows beyond kc and below conf thr.
  for (int k = t; k < POST; k += blockDim.x) {
    float4 o = {0.f, 0.f, 0.f, 0.f};
    if (k < kc) {
      int i = kidx[k];
      float s = sscore[(size_t)b * PPRE + i];
      if (s > 0.5f)
        o = reinterpret_cast<const float4*>(sbox)[(size_t)b * PPRE + i];
    }
    reinterpret_cast<float4*>(out)[(size_t)b * POST + k] = o;
  }
}

// ---------------- host: carve workspace, launch pipeline ----------------
extern "C" void kernel_launch(void* const* d_in, const int* in_sizes, int n_in,
                              void* d_out, int out_size, void* d_ws,
                              size_t ws_size, hipStream_t stream) {
  const float* preds = (const float*)d_in[0];
  const float* regs = (const float*)d_in[1];
  const int* img_h = (const int*)d_in[2];
  const int* img_w = (const int*)d_in[3];
  float* out = (float*)d_out;

  char* w = (char*)d_ws;
  size_t off = 0;
  auto carve = [&](size_t bytes) -> char* {
    char* p = w + off;
    off += (bytes + 255) & ~(size_t)255;
    return p;
  };
  unsigned long long* keys =
      (unsigned long long*)carve((size_t)BS * NPAD * sizeof(unsigned long long));
  float* boxes_all = (float*)carve((size_t)BS * NN * 4 * sizeof(float));
  float* scores_all = (float*)carve((size_t)BS * NN * sizeof(float));
  float* sbox = (float*)carve((size_t)BS * PPRE * 4 * sizeof(float));
  float* sscore = (float*)carve((size_t)BS * PPRE * sizeof(float));
  float* sA = (float*)carve((size_t)BS * PPRE * sizeof(float));
  unsigned short* mask =
      (unsigned short*)carve((size_t)BS * PPRE * PT * sizeof(unsigned short));
  // total ws use ~21.3 MB

  const int NT = 256;
  decode_kernel<<<(BS * NPAD + NT - 1) / NT, NT, 0, stream>>>(
      preds, regs, img_h, img_w, boxes_all, scores_all, keys);

  for (int k = 2; k <= NPAD; k <<= 1)
    for (int j = k >> 1; j > 0; j >>= 1)
      bitonic_kernel<<<(BS * NPAD + NT - 1) / NT, NT, 0, stream>>>(keys, k, j);

  gather_kernel<<<(BS * PPRE + NT - 1) / NT, NT, 0, stream>>>(
      keys, boxes_all, scores_all, sbox, sscore, sA);

  nms_mask_kernel<<<dim3(PPRE / 16, PT / 8, BS), NT, 0, stream>>>(sbox, sA, mask);

  nms_sweep_kernel<<<BS, NT, 0, stream>>>(mask, sbox, sscore, out);
}